// CMA_52956946760162
// MI455X (gfx1250) — compile-verified
//
#include <hip/hip_runtime.h>
#include <hip/hip_bf16.h>

// Problem constants (match reference)
#define CC 4096
#define KK 6
#define DD 2048
#define NN 16384
#define SIGMA 0.2f

typedef __attribute__((ext_vector_type(2))) float v2f;
typedef __attribute__((ext_vector_type(8))) float v8f;

// ---------------------------------------------------------------------------
// Kernel 1: per-segment counts via integer atomics (cheap: 2*N atomics)
// ---------------------------------------------------------------------------
__global__ void cma_count_kernel(const int* __restrict__ labels,
                                 const int* __restrict__ cams,
                                 int* __restrict__ gcnt, int* __restrict__ ccnt,
                                 int n) {
    int i = blockIdx.x * blockDim.x + threadIdx.x;
    if (i < n) {
        int l = labels[i];
        int c = cams[i];
        atomicAdd(&gcnt[l], 1);
        atomicAdd(&ccnt[l * KK + c], 1);
    }
}

// ---------------------------------------------------------------------------
// Kernel 2: single-block exclusive scan (n <= ~32K)
// ---------------------------------------------------------------------------
__global__ __launch_bounds__(1024)
void cma_scan_excl_kernel(const int* __restrict__ cnt, int* __restrict__ offs, int n) {
    __shared__ int part[1024];
    const int tid = threadIdx.x;
    const int chunk = (n + 1023) >> 10;
    const int start = tid * chunk;
    const int end = min(start + chunk, n);
    int s = 0;
    for (int i = start; i < end; ++i) s += cnt[i];
    part[tid] = s;
    __syncthreads();
    for (int off = 1; off < 1024; off <<= 1) {
        int v = (tid >= off) ? part[tid - off] : 0;
        __syncthreads();
        part[tid] += v;
        __syncthreads();
    }
    int base = (tid == 0) ? 0 : part[tid - 1];
    for (int i = start; i < end; ++i) { offs[i] = base; base += cnt[i]; }
}

// ---------------------------------------------------------------------------
// Kernel 3: scatter packed (sample_id << 4) | tile_row into CSR index lists.
// Segment tiles are aligned 16-groups, so tile-local row == seg & 15: no
// per-block row search needed later.
// ---------------------------------------------------------------------------
__global__ void cma_fill_kernel(const int* __restrict__ labels,
                                const int* __restrict__ cams,
                                int* __restrict__ gcur, int* __restrict__ ccur,
                                int* __restrict__ gidx, int* __restrict__ cidx,
                                int n) {
    int i = blockIdx.x * blockDim.x + threadIdx.x;
    if (i < n) {
        int l = labels[i];
        int c = cams[i];
        int seg = l * KK + c;
        int p = atomicAdd(&gcur[l], 1);
        gidx[p] = (i << 4) | (l & 15);
        int q = atomicAdd(&ccur[seg], 1);
        cidx[q] = (i << 4) | (seg & 15);
    }
}

// ---------------------------------------------------------------------------
// Kernel 4: fused WMMA gather segment-sum -> mean -> momentum blend -> output
//   grid = (16 col-groups, nseg/16 segment tiles); block = 256 (8 wave32)
//   A (16x4 f32 one-hot) x B (4x16 f32 gathered feat cols), f32 accumulate
//   via V_WMMA_F32_16X16X4_F32; software-pipelined operand fetch.
// ---------------------------------------------------------------------------
#define STAGE 512

template <bool HAS_VALID>
__global__ __launch_bounds__(256)
void cma_seg_mean_blend_wmma(const float* __restrict__ feats,   // [NN, DD]
                             const int*   __restrict__ idx,     // packed CSR list
                             const int*   __restrict__ offs,    // [nseg] exclusive
                             const int*   __restrict__ cnt,     // [nseg]
                             const float* __restrict__ mem,     // [nseg, DD]
                             const unsigned char* __restrict__ valid, // [nseg]
                             float* __restrict__ outp,          // [nseg, DD]
                             int nseg, int nTotal) {
    const int segbase = blockIdx.y * 16;
    const int tid  = threadIdx.x;
    const int wave = tid >> 5;
    const int lane = tid & 31;
    const int col  = blockIdx.x * 128 + wave * 16 + (lane & 15);

    __shared__ int           s_cnt[16];
    __shared__ unsigned char s_val[16];
    __shared__ int           s_id[STAGE];

    if (tid < 16) {
        s_cnt[tid] = cnt[segbase + tid];
        if (HAS_VALID) s_val[tid] = valid[segbase + tid];
    }

    const int listStart = offs[segbase];
    const int listEnd   = (segbase + 16 < nseg) ? offs[segbase + 16] : nTotal;

    // Hoist the read-once memory-bank row values: stream in (non-temporal)
    // behind the WMMA accumulation work.
    const int mhalf = (lane < 16) ? 0 : 8;
    float mv[8];
    #pragma unroll
    for (int v = 0; v < 8; ++v)
        mv[v] = __builtin_nontemporal_load(&mem[(size_t)(segbase + v + mhalf) * DD + col]);

    v8f acc = {0.f, 0.f, 0.f, 0.f, 0.f, 0.f, 0.f, 0.f};

    const int myrow = lane & 15;
    const int koff  = (lane < 16) ? 0 : 2;   // A/B: VGPR0 holds K=0 (lo half) / K=2 (hi half)

    for (int base = listStart; base < listEnd; base += STAGE) {
        const int nstage = min(STAGE, listEnd - base);
        __syncthreads();   // protect s_id reuse across outer iterations
        for (int p = tid; p < nstage; p += 256) s_id[p] = idx[base + p];
        __syncthreads();

        const int nchunk = (nstage + 3) >> 2;

        // software-pipelined fetch: LDS packed ids + gathered feat columns
        float a0 = 0.f, a1 = 0.f, b0 = 0.f, b1 = 0.f;
        {
            int pk = koff;
            int q0 = min(pk, nstage - 1);
            int q1 = min(pk + 1, nstage - 1);
            int k0 = s_id[q0], k1 = s_id[q1];
            a0 = (pk     < nstage && (k0 & 15) == myrow) ? 1.0f : 0.0f;
            a1 = (pk + 1 < nstage && (k1 & 15) == myrow) ? 1.0f : 0.0f;
            b0 = feats[(size_t)(k0 >> 4) * DD + col];
            b1 = feats[(size_t)(k1 >> 4) * DD + col];
        }
        for (int c = 0; c < nchunk; ++c) {   // uniform trip count: EXEC all-ones
            float na0 = 0.f, na1 = 0.f, nb0 = 0.f, nb1 = 0.f;
            if (c + 1 < nchunk) {            // uniform guard
                int pk = (c + 1) * 4 + koff;
                int q0 = min(pk, nstage - 1);
                int q1 = min(pk + 1, nstage - 1);
                int k0 = s_id[q0], k1 = s_id[q1];
                na0 = (pk     < nstage && (k0 & 15) == myrow) ? 1.0f : 0.0f;
                na1 = (pk + 1 < nstage && (k1 & 15) == myrow) ? 1.0f : 0.0f;
                nb0 = feats[(size_t)(k0 >> 4) * DD + col];
                nb1 = feats[(size_t)(k1 >> 4) * DD + col];
            }
            v2f av = {a0, a1};
            v2f bv = {b0, b1};
            acc = __builtin_amdgcn_wmma_f32_16x16x4_f32(
                false, av, false, bv, (short)0, acc, false, false);
            a0 = na0; a1 = na1; b0 = nb0; b1 = nb1;
        }
    }
    __syncthreads();

    // Epilogue: mean, momentum blend, masked select, non-temporal store.
    // C/D layout: VGPR v <-> row M = v + 8*(lane>=16), col N = lane&15.
    #pragma unroll
    for (int v = 0; v < 8; ++v) {
        int m   = v + mhalf;
        int seg = segbase + m;
        float c    = (float)s_cnt[m];
        float mean = acc[v] / fmaxf(c, 1.0f);
        float blended = (1.0f - SIGMA) * mv[v] + SIGMA * mean;
        float r;
        if (HAS_VALID) {  // per-(label,cam): present ? (valid ? blend : raw mean) : keep
            r = (c > 0.f) ? (s_val[m] ? blended : mean) : mv[v];
        } else {          // global bank: present ? blend : keep
            r = (c > 0.f) ? blended : mv[v];
        }
        __builtin_nontemporal_store(r, &outp[(size_t)seg * DD + col]);
    }
}

// ---------------------------------------------------------------------------
// Host launcher
// ---------------------------------------------------------------------------
extern "C" void kernel_launch(void* const* d_in, const int* in_sizes, int n_in,
                              void* d_out, int out_size, void* d_ws, size_t ws_size,
                              hipStream_t stream) {
    const float* vis_mem   = (const float*)d_in[0];
    const float* ir_mem    = (const float*)d_in[1];
    const float* vis_cmem  = (const float*)d_in[2];
    const float* ir_cmem   = (const float*)d_in[3];
    const unsigned char* vis_valid = (const unsigned char*)d_in[4];
    const unsigned char* ir_valid  = (const unsigned char*)d_in[5];
    const float* rgb_feats = (const float*)d_in[6];
    const float* ir_feats  = (const float*)d_in[7];
    const int* rgb_labels  = (const int*)d_in[8];
    const int* ir_labels   = (const int*)d_in[9];
    const int* rgb_cams    = (const int*)d_in[10];
    const int* ir_cams     = (const int*)d_in[11];
    float* out = (float*)d_out;

    const int SEGTOT = (CC + CC * KK);     // 28672 per modality pair of arrays

    // Workspace layout (ints)
    int* w = (int*)d_ws;
    int* gcnt_v  = w;                 w += CC;
    int* ccnt_v  = w;                 w += CC * KK;
    int* gcnt_i  = w;                 w += CC;
    int* ccnt_i  = w;                 w += CC * KK;
    int* offs0   = w;                 // 4 contiguous offset arrays
    int* goffs_v = w;                 w += CC;
    int* coffs_v = w;                 w += CC * KK;
    int* goffs_i = w;                 w += CC;
    int* coffs_i = w;                 w += CC * KK;
    int* cur0    = w;                 // 4 contiguous cursor arrays (copy of offs)
    int* gcur_v  = w;                 w += CC;
    int* ccur_v  = w;                 w += CC * KK;
    int* gcur_i  = w;                 w += CC;
    int* ccur_i  = w;                 w += CC * KK;
    int* gidx_v  = w;                 w += NN;
    int* cidx_v  = w;                 w += NN;
    int* gidx_i  = w;                 w += NN;
    int* cidx_i  = w;                 w += NN;

    // 1. zero the count arrays
    hipMemsetAsync((void*)gcnt_v, 0, (size_t)(2 * SEGTOT) * sizeof(int), stream);

    // 2. counts
    const int CB = 256, CG = (NN + CB - 1) / CB;
    cma_count_kernel<<<CG, CB, 0, stream>>>(rgb_labels, rgb_cams, gcnt_v, ccnt_v, NN);
    cma_count_kernel<<<CG, CB, 0, stream>>>(ir_labels,  ir_cams,  gcnt_i, ccnt_i, NN);

    // 3. exclusive scans
    cma_scan_excl_kernel<<<1, 1024, 0, stream>>>(gcnt_v, goffs_v, CC);
    cma_scan_excl_kernel<<<1, 1024, 0, stream>>>(ccnt_v, coffs_v, CC * KK);
    cma_scan_excl_kernel<<<1, 1024, 0, stream>>>(gcnt_i, goffs_i, CC);
    cma_scan_excl_kernel<<<1, 1024, 0, stream>>>(ccnt_i, coffs_i, CC * KK);

    // 4. cursors = offsets (single d2d copy of the 4 contiguous arrays), then fill
    hipMemcpyAsync((void*)cur0, (const void*)offs0,
                   (size_t)(2 * SEGTOT) * sizeof(int),
                   hipMemcpyDeviceToDevice, stream);
    cma_fill_kernel<<<CG, CB, 0, stream>>>(rgb_labels, rgb_cams, gcur_v, ccur_v,
                                           gidx_v, cidx_v, NN);
    cma_fill_kernel<<<CG, CB, 0, stream>>>(ir_labels,  ir_cams,  gcur_i, ccur_i,
                                           gidx_i, cidx_i, NN);

    // 5. fused WMMA segment-mean + blend, straight into the packed output
    //    out = [ vg (C,D) | vc (C*K,D) | ig (C,D) | ic (C*K,D) ]
    float* o_vg = out;
    float* o_vc = out + (size_t)CC * DD;
    float* o_ig = out + (size_t)CC * (1 + KK) * DD;
    float* o_ic = out + (size_t)CC * (2 + KK) * DD;

    dim3 blk(256);
    dim3 grid_g(DD / 128, CC / 16);           // (16, 256)
    dim3 grid_c(DD / 128, (CC * KK) / 16);    // (16, 1536)

    cma_seg_mean_blend_wmma<false><<<grid_g, blk, 0, stream>>>(
        rgb_feats, gidx_v, goffs_v, gcnt_v, vis_mem, nullptr, o_vg, CC, NN);
    cma_seg_mean_blend_wmma<true><<<grid_c, blk, 0, stream>>>(
        rgb_feats, cidx_v, coffs_v, ccnt_v, vis_cmem, vis_valid, o_vc, CC * KK, NN);
    cma_seg_mean_blend_wmma<false><<<grid_g, blk, 0, stream>>>(
        ir_feats, gidx_i, goffs_i, gcnt_i, ir_mem, nullptr, o_ig, CC, NN);
    cma_seg_mean_blend_wmma<true><<<grid_c, blk, 0, stream>>>(
        ir_feats, cidx_i, coffs_i, ccnt_i, ir_cmem, ir_valid, o_ic, CC * KK, NN);
}